// SparseBasicBlock_8572754722933
// MI455X (gfx1250) — compile-verified
//
#include <hip/hip_runtime.h>

#define C    128
#define TM   64      // gathered rows per block (4 WMMA M-tiles)
#define PAD  132     // LDS row pad (floats): keeps float4 alignment, dodges bank conflicts
#define EPSB 1e-4f

typedef __attribute__((ext_vector_type(2))) float v2f;
typedef __attribute__((ext_vector_type(8))) float v8f;

// ---------------- zero fill (graph-safe memset) ----------------
__global__ __launch_bounds__(256) void zero_f4(float4* __restrict__ p, long n4) {
    long i = (long)blockIdx.x * blockDim.x + threadIdx.x;
    long stride = (long)gridDim.x * blockDim.x;
    float4 z = {0.f, 0.f, 0.f, 0.f};
    for (; i < n4; i += stride) p[i] = z;
}

// ---------------- submanifold conv: gather -> WMMA GEMM -> scatter-add ----------------
__global__ __launch_bounds__(256) void subm_conv_wmma(
        const float* __restrict__ feat,     // [n][C]
        const float* __restrict__ W,        // [K][C][C] (cin, cout)
        const int*   __restrict__ in_idx,   // [K][n]
        const int*   __restrict__ out_idx,  // [K][n]
        const int*   __restrict__ mask,     // [K][n], 0/1
        float*       __restrict__ out,      // [n][C], accumulated via atomics
        int n)
{
    __shared__ float sA[TM][PAD];   // gathered, masked input rows
    __shared__ float sW[C][PAD];    // W[k], row-major (cin, cout)
    __shared__ int   sOut[TM];
    __shared__ int   sMask[TM];

    const int  k       = blockIdx.y;
    const long rowBase = (long)blockIdx.x * TM;
    const int  tid     = threadIdx.x;
    const int  lane    = tid & 31;
    const int  wave    = tid >> 5;          // 0..7 -> 16-col N-tile

    // ---- stage W[k] into LDS (float4, fully coalesced) ----
    const float4* Wk4 = (const float4*)(W + (size_t)k * C * C);
    for (int i = tid; i < C * C / 4; i += 256) {
        float4 w  = Wk4[i];
        int cin   = (i * 4) >> 7;
        int cout  = (i * 4) & (C - 1);
        *(float4*)&sW[cin][cout] = w;
    }

    // ---- rulebook rows for this tile ----
    if (tid < TM) {
        long row = rowBase + tid;
        int m = 0, oi = 0;
        if (row < n) {
            m  = mask[(size_t)k * n + row];
            oi = out_idx[(size_t)k * n + row];
        }
        sMask[tid] = (row < n) ? m : 0;
        sOut[tid]  = oi;
    }

    // ---- masked gather of 64 input rows (4 threads/row, float4) ----
    {
        const int rpt = 256 / TM;           // 4 threads per row
        const int r   = tid / rpt;
        const int sub = tid % rpt;
        long row = rowBase + r;
        bool live = false; int src = 0;
        if (row < n) {
            int m = mask[(size_t)k * n + row];
            if (m != 0) { live = true; src = in_idx[(size_t)k * n + row]; }
        }
        const float4* srcp = (const float4*)(feat + (size_t)src * C);
        float4 z = {0.f, 0.f, 0.f, 0.f};
        #pragma unroll
        for (int j = 0; j < C / 4 / (256 / TM); ++j) {     // 8 float4 per thread
            int idx4 = sub * (C / 4 / (256 / TM)) + j;
            float4 v = live ? srcp[idx4] : z;              // mask is 0/1 -> zero row
            *(float4*)&sA[r][idx4 * 4] = v;
        }
    }
    __syncthreads();

    // ---- GEMM: 4 M-tiles x 1 N-tile per wave, K=128 in steps of 4 (fp32 WMMA) ----
    v8f acc0 = {}, acc1 = {}, acc2 = {}, acc3 = {};
    const int half = lane >> 4;   // 0: lanes 0-15, 1: lanes 16-31
    const int l16  = lane & 15;
    const int nb   = wave * 16;

    #pragma unroll 4
    for (int s = 0; s < C / 4; ++s) {
        const int kb = s * 4;
        // B fragment (4x16): VGPR v holds K=v (lanes 0-15) / K=v+2 (lanes 16-31)
        v2f b;
        b.x = sW[kb + 2 * half + 0][nb + l16];
        b.y = sW[kb + 2 * half + 1][nb + l16];
        // A fragment (16x4): lane=M, VGPR v holds K = 2*half + v  -> contiguous pair
        v2f a0 = *(const v2f*)&sA[ 0 + l16][kb + 2 * half];
        v2f a1 = *(const v2f*)&sA[16 + l16][kb + 2 * half];
        v2f a2 = *(const v2f*)&sA[32 + l16][kb + 2 * half];
        v2f a3 = *(const v2f*)&sA[48 + l16][kb + 2 * half];
        acc0 = __builtin_amdgcn_wmma_f32_16x16x4_f32(false, a0, false, b, (short)0, acc0, false, false);
        acc1 = __builtin_amdgcn_wmma_f32_16x16x4_f32(false, a1, false, b, (short)0, acc1, false, false);
        acc2 = __builtin_amdgcn_wmma_f32_16x16x4_f32(false, a2, false, b, (short)0, acc2, false, false);
        acc3 = __builtin_amdgcn_wmma_f32_16x16x4_f32(false, a3, false, b, (short)0, acc3, false, false);
    }

    // ---- scatter-add: D vgpr r -> M = r + 8*half, N = l16; skip masked-out rows ----
    #pragma unroll
    for (int m = 0; m < 4; ++m) {
        v8f acc = (m == 0) ? acc0 : (m == 1) ? acc1 : (m == 2) ? acc2 : acc3;
        #pragma unroll
        for (int r = 0; r < 8; ++r) {
            int lrow = m * 16 + r + half * 8;
            if (sMask[lrow] != 0) {
                long grow = sOut[lrow];
                atomicAdd(&out[grow * (long)C + nb + l16], acc[r]);
            }
        }
    }
}

// ---------------- BN + ReLU, in place ----------------
__global__ __launch_bounds__(256) void bn_relu_inplace(
        float* __restrict__ h, const float* __restrict__ g, const float* __restrict__ bt,
        const float* __restrict__ mn, const float* __restrict__ vr, long total4)
{
    __shared__ float ss[C], sb[C];
    if (threadIdx.x < C) {
        float s = g[threadIdx.x] * rsqrtf(vr[threadIdx.x] + EPSB);
        ss[threadIdx.x] = s;
        sb[threadIdx.x] = bt[threadIdx.x] - mn[threadIdx.x] * s;
    }
    __syncthreads();
    float4* p = (float4*)h;
    long i = (long)blockIdx.x * blockDim.x + threadIdx.x;
    long stride = (long)gridDim.x * blockDim.x;
    for (; i < total4; i += stride) {
        float4 v = p[i];
        int c = (int)((i * 4) & (C - 1));
        v.x = fmaxf(v.x * ss[c + 0] + sb[c + 0], 0.f);
        v.y = fmaxf(v.y * ss[c + 1] + sb[c + 1], 0.f);
        v.z = fmaxf(v.z * ss[c + 2] + sb[c + 2], 0.f);
        v.w = fmaxf(v.w * ss[c + 3] + sb[c + 3], 0.f);
        p[i] = v;
    }
}

// ---------------- BN2 + residual add, in place on d_out ----------------
__global__ __launch_bounds__(256) void bn_add_residual(
        float* __restrict__ h, const float* __restrict__ feat,
        const float* __restrict__ g, const float* __restrict__ bt,
        const float* __restrict__ mn, const float* __restrict__ vr, long total4)
{
    __shared__ float ss[C], sb[C];
    if (threadIdx.x < C) {
        float s = g[threadIdx.x] * rsqrtf(vr[threadIdx.x] + EPSB);
        ss[threadIdx.x] = s;
        sb[threadIdx.x] = bt[threadIdx.x] - mn[threadIdx.x] * s;
    }
    __syncthreads();
    float4* p = (float4*)h;
    const float4* f = (const float4*)feat;
    long i = (long)blockIdx.x * blockDim.x + threadIdx.x;
    long stride = (long)gridDim.x * blockDim.x;
    for (; i < total4; i += stride) {
        float4 v = p[i];
        float4 r = f[i];
        int c = (int)((i * 4) & (C - 1));
        v.x = v.x * ss[c + 0] + sb[c + 0] + r.x;
        v.y = v.y * ss[c + 1] + sb[c + 1] + r.y;
        v.z = v.z * ss[c + 2] + sb[c + 2] + r.z;
        v.w = v.w * ss[c + 3] + sb[c + 3] + r.w;
        p[i] = v;
    }
}

extern "C" void kernel_launch(void* const* d_in, const int* in_sizes, int n_in,
                              void* d_out, int out_size, void* d_ws, size_t ws_size,
                              hipStream_t stream) {
    const float* feat   = (const float*)d_in[0];
    const float* W1     = (const float*)d_in[1];
    const float* W2     = (const float*)d_in[2];
    const float* gamma1 = (const float*)d_in[3];
    const float* beta1  = (const float*)d_in[4];
    const float* mean1  = (const float*)d_in[5];
    const float* var1   = (const float*)d_in[6];
    const float* gamma2 = (const float*)d_in[7];
    const float* beta2  = (const float*)d_in[8];
    const float* mean2  = (const float*)d_in[9];
    const float* var2   = (const float*)d_in[10];
    const int* in_idx   = (const int*)d_in[11];
    const int* out_idx  = (const int*)d_in[12];
    const int* mask     = (const int*)d_in[13];

    const int  n      = in_sizes[0] / C;
    float*     out    = (float*)d_out;
    float*     h1     = (float*)d_ws;          // N*C fp32 intermediate
    const long total4 = (long)n * C / 4;

    dim3 cgrid((n + TM - 1) / TM, 4);

    // conv1 -> h1 (zero, scatter-accumulate), BN1+ReLU in place
    zero_f4<<<4096, 256, 0, stream>>>((float4*)h1, total4);
    subm_conv_wmma<<<cgrid, 256, 0, stream>>>(feat, W1, in_idx, out_idx, mask, h1, n);
    bn_relu_inplace<<<8192, 256, 0, stream>>>(h1, gamma1, beta1, mean1, var1, total4);

    // conv2 -> d_out (zero, scatter-accumulate), then BN2 + residual in place
    zero_f4<<<4096, 256, 0, stream>>>((float4*)out, total4);
    subm_conv_wmma<<<cgrid, 256, 0, stream>>>(h1, W2, in_idx, out_idx, mask, out, n);
    bn_add_residual<<<8192, 256, 0, stream>>>(out, feat, gamma2, beta2, mean2, var2, total4);
}